// AngularPenaltySMLoss_7370163880561
// MI455X (gfx1250) — compile-verified
//
#include <hip/hip_runtime.h>
#include <cstdint>
#include <cstddef>

// ArcFace loss, N=4096 rows x C=32000 classes, f32, scalar output.
// HBM-bound: 524 MB single pass -> ~22.5 us floor @ 23.3 TB/s.
// Streams rows through LDS with gfx1250 async global->LDS loads.

#define S_SCALE 64.0f
#define COS_M   0.8775825618903728f   // cos(0.5)
#define SIN_M   0.479425538604203f    // sin(0.5)
#define CLIP_EPS 1e-5f

constexpr int TPB    = 256;           // 8 waves (wave32)
constexpr int VEC    = 4;             // floats per lane per stage (b128)
constexpr int TILE   = TPB * VEC;     // 1024 floats / 4 KB per stage
constexpr int STAGES = 8;             // in-flight stages per wave -> 32 KB LDS
constexpr int NWARP  = TPB / 32;

__device__ __forceinline__ void async_issue_b128(const float* gptr, unsigned lds_byte_off) {
  // Per-lane async copy of 16B global -> LDS. Tracked by ASYNCcnt (not LOADcnt).
  asm volatile("global_load_async_to_lds_b128 %0, %1, off"
               :: "v"(lds_byte_off), "v"(gptr)
               : "memory");
}

__global__ __launch_bounds__(TPB) void arcface_row_kernel(
    const float* __restrict__ x,
    const int*  __restrict__ labels,
    float* __restrict__ row_loss,
    int C) {
  __shared__ float tile[STAGES * TILE];   // 32 KB staging buffer
  __shared__ float warp_part[NWARP];

  const int row = blockIdx.x;
  const int tid = threadIdx.x;
  const float* rowp = x + (size_t)row * (size_t)C;

  // true-class logit (single lane; tiny vs the 128 KB row stream)
  float true_logit = 0.0f;
  if (tid == 0) true_logit = rowp[labels[row]];

  // workgroup-relative LDS byte offset of the staging buffer
  const unsigned lds0 = (unsigned)(uintptr_t)(&tile[0]);
  const unsigned lane_off = (unsigned)(tid * VEC) * 4u;

  const int num_tiles = (C + TILE - 1) / TILE;

  // ---- prologue: fill the async pipeline -------------------------------
  const int pre = num_tiles < STAGES ? num_tiles : STAGES;
  for (int t = 0; t < pre; ++t) {
    const int elem = t * TILE + tid * VEC;
    if (elem < C)   // C % 4 == 0 -> vec4 chunk is all-in or all-out
      async_issue_b128(rowp + elem, lds0 + (unsigned)(t % STAGES) * (TILE * 4u) + lane_off);
  }

  float acc = 0.0f;
  int t = 0;
  const int steady = num_tiles - STAGES;   // all tiles in this region are full

  // ---- steady state: wait oldest, consume, refill same slot ------------
  for (; t < steady; ++t) {
    // async loads complete in order per wave: <=7 outstanding => stage t landed
    asm volatile("s_wait_asynccnt %0" :: "n"(STAGES - 1) : "memory");
    const int slot = t % STAGES;
    const float* lp = &tile[slot * TILE + tid * VEC];
    const float v0 = lp[0], v1 = lp[1], v2 = lp[2], v3 = lp[3];
    acc += __expf(S_SCALE * v0);
    acc += __expf(S_SCALE * v1);
    acc += __expf(S_SCALE * v2);
    acc += __expf(S_SCALE * v3);
    // async LDS writes are unordered vs ds_loads: drain DS before recycling slot
    asm volatile("s_wait_dscnt 0" ::: "memory");
    const int nelem = (t + STAGES) * TILE + tid * VEC;
    if (nelem < C)
      async_issue_b128(rowp + nelem, lds0 + (unsigned)slot * (TILE * 4u) + lane_off);
  }

  // ---- drain: everything issued, wait for all, consume the rest --------
  asm volatile("s_wait_asynccnt 0" ::: "memory");
  for (; t < num_tiles; ++t) {
    const int slot = t % STAGES;
    const int elem = t * TILE + tid * VEC;
    if (elem < C) {
      const float* lp = &tile[slot * TILE + tid * VEC];
      acc += __expf(S_SCALE * lp[0]);
      acc += __expf(S_SCALE * lp[1]);
      acc += __expf(S_SCALE * lp[2]);
      acc += __expf(S_SCALE * lp[3]);
    }
  }

  // ---- block reduction (wave32 shuffles, then LDS partials) ------------
  for (int m = 16; m >= 1; m >>= 1) acc += __shfl_xor(acc, m, 32);
  if ((tid & 31) == 0) warp_part[tid >> 5] = acc;
  __syncthreads();

  if (tid == 0) {
    float sum = 0.0f;
    for (int w = 0; w < NWARP; ++w) sum += warp_part[w];
    const float tl = true_logit;
    const float c  = fminf(fmaxf(tl, -1.0f + CLIP_EPS), 1.0f - CLIP_EPS);
    // cos(acos(c) + M) = c*cosM - sqrt(1-c^2)*sinM   (sin(acos(c)) >= 0)
    const float num   = S_SCALE * (c * COS_M - sqrtf(fmaxf(1.0f - c * c, 0.0f)) * SIN_M);
    const float denom = expf(num) + (sum - expf(S_SCALE * tl));
    row_loss[row] = logf(denom) - num;   // == -L_i
  }
}

__global__ __launch_bounds__(256) void arcface_final_reduce(
    const float* __restrict__ row_loss, float* __restrict__ out, int N) {
  __shared__ float part[8];
  float acc = 0.0f;
  for (int i = threadIdx.x; i < N; i += 256) acc += row_loss[i];
  for (int m = 16; m >= 1; m >>= 1) acc += __shfl_xor(acc, m, 32);
  if ((threadIdx.x & 31) == 0) part[threadIdx.x >> 5] = acc;
  __syncthreads();
  if (threadIdx.x == 0) {
    float s = 0.0f;
    for (int w = 0; w < 8; ++w) s += part[w];
    out[0] = s / (float)N;   // -mean(L)
  }
}

extern "C" void kernel_launch(void* const* d_in, const int* in_sizes, int n_in,
                              void* d_out, int out_size, void* d_ws, size_t ws_size,
                              hipStream_t stream) {
  const float* x      = (const float*)d_in[0];
  const int*   labels = (const int*)d_in[1];
  const int N = in_sizes[1];
  const int C = in_sizes[0] / N;
  float* row_loss = (float*)d_ws;          // N floats of scratch

  arcface_row_kernel<<<N, TPB, 0, stream>>>(x, labels, row_loss, C);
  arcface_final_reduce<<<1, 256, 0, stream>>>(row_loss, (float*)d_out, N);
}